// VDPWIModel_91156385890925
// MI455X (gfx1250) — compile-verified
//
#include <hip/hip_runtime.h>
#include <math.h>

typedef __attribute__((ext_vector_type(2))) float v2f;
typedef __attribute__((ext_vector_type(8))) float v8f;

#define LSEQ 32
#define EMB  300
#define HID  250
#define G4   1000   // 4*HID
#define NLAB 5

// ---- workspace layout (float offsets) ----
#define WS_XPROJ 0        // 2 * 32*1000 = 64000
#define WS_S     64000    // 2 * 32*500  = 32000
#define WS_GRAM  96000    // 4 * 32*32   = 4096
#define WS_NORM  100096   // 2 * 4 * 32  = 256
#define WS_SIM   100352   // 12*32*32    = 12288
#define WS_A0    112640   // 12*32*32    = 12288
#define WS_BUFA  124928   // 32768
#define WS_BUFB  157696   // 32768
// total ~190464 floats (~762 KB)

// --------------------------------------------------------------------------
// Kernel A: gather embeddings + input projection GEMM (fp32 WMMA 16x16x4)
//   xproj[seq][t][o] = emb[x[t]] . w_ih[o] + b_ih[o] + b_hh[o]
//   One 16x16 tile per 32-thread block. Grid = 2 seq * 2 Mtiles * 63 Ntiles.
//   Out-of-range N lanes use clamped addresses + x0 multiplier (no divergent
//   branches in the WMMA loop -> EXEC stays all-ones throughout).
// --------------------------------------------------------------------------
__global__ void xproj_kernel(const int* __restrict__ x1, const int* __restrict__ x2,
                             const float* __restrict__ emb, const float* __restrict__ w_ih,
                             const float* __restrict__ b_ih, const float* __restrict__ b_hh,
                             float* __restrict__ ws) {
    int blk = blockIdx.x;            // 0..251
    int seq = blk / 126;
    int rem = blk % 126;
    int mt  = rem / 63;
    int nt  = rem % 63;
    int lane = threadIdx.x & 31;
    int half = lane >> 4;            // 0/1
    int l15  = lane & 15;
    int kb   = half * 2;

    const int* xs = seq ? x2 : x1;
    int  m    = mt * 16 + l15;                 // A-fragment row (always < 32)
    long erow = (long)xs[m] * EMB;

    int   n    = nt * 16 + l15;                // B-fragment col (may be >= 1000)
    int   ncl  = (n < G4) ? n : (G4 - 1);      // clamped, always valid
    float bsel = (n < G4) ? 1.0f : 0.0f;       // select multiplier

    // fold biases into accumulator init (bias depends only on n == C/D col)
    v8f acc;
    float bias = (b_ih[ncl] + b_hh[ncl]) * bsel;
    for (int r = 0; r < 8; ++r) acc[r] = bias;

    const float* ea = emb + erow;
    const float* wr = w_ih + (long)ncl * EMB;

    for (int k = 0; k < EMB; k += 4) {         // EMB=300, multiple of 4
        v2f a, b;
        a[0] = ea[k + kb];
        a[1] = ea[k + kb + 1];
        b[0] = wr[k + kb]     * bsel;
        b[1] = wr[k + kb + 1] * bsel;
        acc = __builtin_amdgcn_wmma_f32_16x16x4_f32(
            false, a, false, b, (short)0, acc, false, false);
    }

    float* xp = ws + WS_XPROJ + seq * (LSEQ * G4);
    if (n < G4) {
        for (int r = 0; r < 8; ++r) {
            int mc = mt * 16 + half * 8 + r;
            xp[mc * G4 + n] = acc[r];
        }
    }
}

// --------------------------------------------------------------------------
// Kernel B: 4 independent LSTM chains (seq x direction), one block each.
//   Sequential 32-step recurrence; per step a 1000x250 GEMV against h.
// --------------------------------------------------------------------------
__global__ void lstm_kernel(const float* __restrict__ w_hh, float* __restrict__ ws) {
    int cid = blockIdx.x;        // 0..3
    int seq = cid >> 1;
    int dir = cid & 1;           // 0 = forward, 1 = backward
    int tid = threadIdx.x;       // 0..255

    __shared__ float h[HID];
    __shared__ float cst[HID];
    __shared__ float g[G4];

    if (tid < HID) { h[tid] = 0.0f; cst[tid] = 0.0f; }
    __syncthreads();

    const float* xp   = ws + WS_XPROJ + seq * (LSEQ * G4);
    float*       sout = ws + WS_S     + seq * (LSEQ * 2 * HID);

    for (int t = 0; t < LSEQ; ++t) {
        int row = dir ? (LSEQ - 1 - t) : t;
        const float* xr = xp + row * G4;
        if (t + 1 < LSEQ) {   // hint next step's row toward cache
            int nrow = dir ? (LSEQ - 2 - t) : (t + 1);
            __builtin_prefetch(xp + nrow * G4, 0, 1);
        }
        // phase 1: gate pre-activations  g[o] = xproj + h . w_hh[o]
        for (int o = tid; o < G4; o += 256) {
            const float* wr = w_hh + (long)o * HID;
            float s = xr[o];
            for (int k = 0; k < HID; ++k) s = fmaf(h[k], wr[k], s);
            g[o] = s;
        }
        __syncthreads();
        // phase 2: elementwise LSTM cell update (gate order i,f,g,o)
        if (tid < HID) {
            float gi = g[tid], gf = g[HID + tid], gg = g[2 * HID + tid], go = g[3 * HID + tid];
            float si = 1.0f / (1.0f + __expf(-gi));
            float sf = 1.0f / (1.0f + __expf(-gf));
            float so = 1.0f / (1.0f + __expf(-go));
            float c  = sf * cst[tid] + si * tanhf(gg);
            float hn = so * tanhf(c);
            cst[tid] = c;
            h[tid]   = hn;
            sout[t * (2 * HID) + dir * HID + tid] = hn;
        }
        __syncthreads();
    }
}

// --------------------------------------------------------------------------
// Kernel C1: four 32x32 Gram matrices (full/fwd/bwd/sum slices) via fp32 WMMA.
//   Grid = 4 groups * 4 (2x2) tiles, 32 threads each.
//   Main K loop is unguarded; the single partial tail (D=250 only) uses
//   clamped addresses + select multipliers so EXEC never diverges.
// --------------------------------------------------------------------------
__global__ void gram_kernel(float* __restrict__ ws) {
    int blk  = blockIdx.x;           // g*4 + mt*2 + nt
    int gidx = blk >> 2;
    int mt   = (blk >> 1) & 1;
    int nt   = blk & 1;
    int lane = threadIdx.x & 31;
    int half = lane >> 4;
    int l15  = lane & 15;
    int kb   = half * 2;

    const float* s1 = ws + WS_S;
    const float* s2 = ws + WS_S + LSEQ * 2 * HID;

    int  D     = (gidx == 0) ? (2 * HID) : HID;
    int  coff  = (gidx == 2) ? HID : 0;
    bool issum = (gidx == 3);

    const float* arow = s1 + (mt * 16 + l15) * (2 * HID) + coff;
    const float* brow = s2 + (nt * 16 + l15) * (2 * HID) + coff;

    v8f acc;
    for (int r = 0; r < 8; ++r) acc[r] = 0.0f;

    int kmain = D & ~3;              // 500 -> 500, 250 -> 248
    if (issum) {
        for (int k = 0; k < kmain; k += 4) {
            int k0 = k + kb, k1 = k0 + 1;
            v2f a, b;
            a[0] = arow[k0] + arow[HID + k0];
            a[1] = arow[k1] + arow[HID + k1];
            b[0] = brow[k0] + brow[HID + k0];
            b[1] = brow[k1] + brow[HID + k1];
            acc = __builtin_amdgcn_wmma_f32_16x16x4_f32(
                false, a, false, b, (short)0, acc, false, false);
        }
    } else {
        for (int k = 0; k < kmain; k += 4) {
            int k0 = k + kb, k1 = k0 + 1;
            v2f a, b;
            a[0] = arow[k0];
            a[1] = arow[k1];
            b[0] = brow[k0];
            b[1] = brow[k1];
            acc = __builtin_amdgcn_wmma_f32_16x16x4_f32(
                false, a, false, b, (short)0, acc, false, false);
        }
    }
    if (kmain < D) {                 // tail: only D=250 (k=248; half1 lanes OOB)
        int k0 = kmain + kb, k1 = k0 + 1;
        float s0 = (k0 < D) ? 1.0f : 0.0f;
        float s1v = (k1 < D) ? 1.0f : 0.0f;
        int  c0 = (k0 < D) ? k0 : (D - 1);
        int  c1 = (k1 < D) ? k1 : (D - 1);
        float a0 = arow[c0], a1 = arow[c1], b0 = brow[c0], b1 = brow[c1];
        if (issum) {
            a0 += arow[HID + c0]; a1 += arow[HID + c1];
            b0 += brow[HID + c0]; b1 += brow[HID + c1];
        }
        v2f a, b;
        a[0] = a0 * s0; a[1] = a1 * s1v;
        b[0] = b0 * s0; b[1] = b1 * s1v;
        acc = __builtin_amdgcn_wmma_f32_16x16x4_f32(
            false, a, false, b, (short)0, acc, false, false);
    }

    float* gm = ws + WS_GRAM + gidx * (32 * 32);
    for (int r = 0; r < 8; ++r) {
        int mc = mt * 16 + half * 8 + r;
        gm[mc * 32 + l15 + nt * 16] = acc[r];
    }
}

// --------------------------------------------------------------------------
// Kernel C2: row norms for the 4 slice groups x 2 sequences x 32 rows.
// --------------------------------------------------------------------------
__global__ void norm_kernel(float* __restrict__ ws) {
    int tid  = threadIdx.x;          // 0..255
    int seqi = tid >> 7;
    int gidx = (tid >> 5) & 3;
    int row  = tid & 31;
    const float* s = ws + WS_S + seqi * (LSEQ * 2 * HID) + row * (2 * HID);
    float acc = 0.0f;
    if (gidx == 0)      { for (int k = 0; k < 2 * HID; ++k) acc = fmaf(s[k], s[k], acc); }
    else if (gidx == 1) { for (int k = 0; k < HID; ++k)     acc = fmaf(s[k], s[k], acc); }
    else if (gidx == 2) { for (int k = 0; k < HID; ++k)     acc = fmaf(s[HID + k], s[HID + k], acc); }
    else { for (int k = 0; k < HID; ++k) { float v = s[k] + s[HID + k]; acc = fmaf(v, v, acc); } }
    ws[WS_NORM + seqi * 128 + gidx * 32 + row] = sqrtf(acc);
}

// --------------------------------------------------------------------------
// Kernel C3: assemble sim cube [12][32][32] from Gram + norms.
//   ch = 3g+0: dot, 3g+1: cosine, 3g+2: L2 (via ||a||^2+||b||^2-2dot).
// --------------------------------------------------------------------------
__global__ void sim_kernel(float* __restrict__ ws) {
    int idx = blockIdx.x * blockDim.x + threadIdx.x;
    if (idx >= 12 * 32 * 32) return;
    int ch = idx >> 10;
    int i  = (idx >> 5) & 31;
    int j  = idx & 31;
    int g  = ch / 3, t = ch % 3;
    float dot = ws[WS_GRAM + g * 1024 + i * 32 + j];
    float na  = ws[WS_NORM + g * 32 + i];
    float nb  = ws[WS_NORM + 128 + g * 32 + j];
    float v;
    if (t == 0)      v = dot;
    else if (t == 1) v = dot / (na * nb + 1e-8f);
    else             v = sqrtf(fmaxf(na * na + nb * nb - 2.0f * dot, 0.0f));
    ws[WS_SIM + idx] = v;
}

// --------------------------------------------------------------------------
// Kernel D: greedy one-to-one matching on sim[9] and sim[10] (OR of masks),
//   then focus-weighted copy sim -> a0. Single wave32; winner selection via
//   __shfl_xor reduction; all state in registers (replicated bitmasks).
// --------------------------------------------------------------------------
__global__ void greedy_kernel(float* __restrict__ ws) {
    int lane = threadIdx.x & 31;     // lane == row it owns
    unsigned selMask = 0u;           // selected cols in this lane's row

    for (int chpass = 0; chpass < 2; ++chpass) {
        int ch = (chpass == 0) ? 9 : 10;
        const float* sm = ws + WS_SIM + ch * 1024;
        unsigned picked = 0u;                 // this row's already-picked entries
        unsigned rowUsed = 0u, colUsed = 0u;  // identical on every lane
        for (int it = 0; it < 64; ++it) {     // k = min(n1+n2, n1*n2) = 64
            float bv = -__builtin_inff();
            int bflat = 0x7fffffff;
            for (int j = 0; j < 32; ++j) {
                if (!((picked >> j) & 1u)) {
                    float v = sm[lane * 32 + j];
                    int  fl = lane * 32 + j;
                    if (v > bv || (v == bv && fl < bflat)) { bv = v; bflat = fl; }
                }
            }
            // wave-wide argmax (ties -> lowest flat index, matching top_k order)
            for (int off = 16; off > 0; off >>= 1) {
                float ov  = __shfl_xor(bv, off, 32);
                int   ofl = __shfl_xor(bflat, off, 32);
                if (ov > bv || (ov == bv && ofl < bflat)) { bv = ov; bflat = ofl; }
            }
            int r = bflat >> 5, c = bflat & 31;
            if (lane == r) picked |= (1u << c);
            bool ok = !((rowUsed >> r) & 1u) && !((colUsed >> c) & 1u);
            if (ok) {
                rowUsed |= (1u << r);
                colUsed |= (1u << c);
                if (lane == r) selMask |= (1u << c);
            }
        }
    }
    // focus = where(sel, 1.0, 0.1) * sim  ->  CNN input a0[12][32][32]
    for (int ch = 0; ch < 12; ++ch) {
        for (int j = 0; j < 32; ++j) {
            float v = ws[WS_SIM + ch * 1024 + lane * 32 + j];
            float w = ((selMask >> j) & 1u) ? 1.0f : 0.1f;
            ws[WS_A0 + ch * 1024 + lane * 32 + j] = v * w;
        }
    }
}

// --------------------------------------------------------------------------
// Conv 3x3 SAME + ReLU + 2x2 maxpool, fused: one thread per pooled output.
// --------------------------------------------------------------------------
__global__ void convpool_kernel(const float* __restrict__ in, const float* __restrict__ w,
                                const float* __restrict__ bias, float* __restrict__ out,
                                int Cin, int Hin, int Cout) {
    int Hp = Hin >> 1;
    int total = Cout * Hp * Hp;
    int idx = blockIdx.x * blockDim.x + threadIdx.x;
    if (idx >= total) return;
    int co = idx / (Hp * Hp);
    int p  = idx % (Hp * Hp);
    int ph = p / Hp, pw = p % Hp;
    const float* wc = w + (long)co * Cin * 9;
    float best = 0.0f;                         // relu(x) >= 0, full pool window
    for (int dy = 0; dy < 2; ++dy)
    for (int dx = 0; dx < 2; ++dx) {
        int oh = 2 * ph + dy, ow = 2 * pw + dx;
        float s = bias[co];
        for (int ci = 0; ci < Cin; ++ci) {
            const float* wp = wc + ci * 9;
            for (int ky = 0; ky < 3; ++ky) {
                int ih = oh + ky - 1;
                if (ih < 0 || ih >= Hin) continue;
                const float* ip = in + ((long)ci * Hin + ih) * Hin;
                for (int kx = 0; kx < 3; ++kx) {
                    int iw = ow + kx - 1;
                    if (iw < 0 || iw >= Hin) continue;
                    s = fmaf(ip[iw], wp[ky * 3 + kx], s);
                }
            }
        }
        best = fmaxf(best, fmaxf(s, 0.0f));
    }
    out[idx] = best;
}

// --------------------------------------------------------------------------
// Head: 128->128 ReLU dense then 128->5.
// --------------------------------------------------------------------------
__global__ void head_kernel(const float* __restrict__ dnn_w, const float* __restrict__ dnn_b,
                            const float* __restrict__ out_w, const float* __restrict__ out_b,
                            float* __restrict__ ws, float* __restrict__ dout) {
    __shared__ float hid[128];
    int tid = threadIdx.x;                     // 0..127
    const float* flat = ws + WS_BUFA;          // layer-5 output [128]
    float s = dnn_b[tid];
    for (int k = 0; k < 128; ++k) s = fmaf(flat[k], dnn_w[tid * 128 + k], s);
    hid[tid] = fmaxf(s, 0.0f);
    __syncthreads();
    if (tid < NLAB) {
        float o = out_b[tid];
        for (int k = 0; k < 128; ++k) o = fmaf(hid[k], out_w[tid * 128 + k], o);
        dout[tid] = o;
    }
}

// --------------------------------------------------------------------------
extern "C" void kernel_launch(void* const* d_in, const int* in_sizes, int n_in,
                              void* d_out, int out_size, void* d_ws, size_t ws_size,
                              hipStream_t stream) {
    (void)in_sizes; (void)n_in; (void)out_size; (void)ws_size;
    const int*   x1    = (const int*)d_in[0];
    const int*   x2    = (const int*)d_in[1];
    const float* emb   = (const float*)d_in[2];
    const float* w_ih  = (const float*)d_in[3];
    const float* w_hh  = (const float*)d_in[4];
    const float* b_ih  = (const float*)d_in[5];
    const float* b_hh  = (const float*)d_in[6];
    const float* cw1   = (const float*)d_in[7];  const float* cb1 = (const float*)d_in[8];
    const float* cw2   = (const float*)d_in[9];  const float* cb2 = (const float*)d_in[10];
    const float* cw3   = (const float*)d_in[11]; const float* cb3 = (const float*)d_in[12];
    const float* cw4   = (const float*)d_in[13]; const float* cb4 = (const float*)d_in[14];
    const float* cw5   = (const float*)d_in[15]; const float* cb5 = (const float*)d_in[16];
    const float* dnn_w = (const float*)d_in[17]; const float* dnn_b = (const float*)d_in[18];
    const float* out_w = (const float*)d_in[19]; const float* out_b = (const float*)d_in[20];
    float* ws  = (float*)d_ws;
    float* out = (float*)d_out;

    // A: embeddings + input projection (WMMA f32)
    hipLaunchKernelGGL(xproj_kernel, dim3(252), dim3(32), 0, stream,
                       x1, x2, emb, w_ih, b_ih, b_hh, ws);
    // B: 4 parallel LSTM chains
    hipLaunchKernelGGL(lstm_kernel, dim3(4), dim3(256), 0, stream, w_hh, ws);
    // C: similarity cube
    hipLaunchKernelGGL(gram_kernel, dim3(16), dim3(32), 0, stream, ws);
    hipLaunchKernelGGL(norm_kernel, dim3(1), dim3(256), 0, stream, ws);
    hipLaunchKernelGGL(sim_kernel, dim3(48), dim3(256), 0, stream, ws);
    // D: greedy matching + focus weighting
    hipLaunchKernelGGL(greedy_kernel, dim3(1), dim3(32), 0, stream, ws);
    // E: CNN tower (fused conv+relu+pool), ping-pong buffers
    float* a0   = ws + WS_A0;
    float* bufA = ws + WS_BUFA;
    float* bufB = ws + WS_BUFB;
    hipLaunchKernelGGL(convpool_kernel, dim3(128), dim3(256), 0, stream, a0,   cw1, cb1, bufA, 12,  32, 128);
    hipLaunchKernelGGL(convpool_kernel, dim3(41),  dim3(256), 0, stream, bufA, cw2, cb2, bufB, 128, 16, 164);
    hipLaunchKernelGGL(convpool_kernel, dim3(12),  dim3(256), 0, stream, bufB, cw3, cb3, bufA, 164, 8,  192);
    hipLaunchKernelGGL(convpool_kernel, dim3(3),   dim3(256), 0, stream, bufA, cw4, cb4, bufB, 192, 4,  192);
    hipLaunchKernelGGL(convpool_kernel, dim3(1),   dim3(256), 0, stream, bufB, cw5, cb5, bufA, 192, 2,  128);
    // F: dense head -> d_out[5]
    hipLaunchKernelGGL(head_kernel, dim3(1), dim3(128), 0, stream,
                       dnn_w, dnn_b, out_w, out_b, ws, out);
}